// GCN_Examp_19516331393575
// MI455X (gfx1250) — compile-verified
//
#include <hip/hip_runtime.h>
#include <hip/hip_bf16.h>
#include <math.h>

#define N_NODES 100000
#define FEAT_IN 128

typedef __attribute__((ext_vector_type(2))) float v2f;
typedef __attribute__((ext_vector_type(8))) float v8f;

// ---------------------------------------------------------------------------
// Degree / normalization
// ---------------------------------------------------------------------------
__global__ void k_init_deg(float* __restrict__ deg, int n) {
    int i = blockIdx.x * blockDim.x + threadIdx.x;
    if (i < n) deg[i] = 1.0f;  // self-loop contributes 1 to every node's degree
}

__global__ void k_deg_acc(const int* __restrict__ dst, float* __restrict__ deg, int e) {
    int i = blockIdx.x * blockDim.x + threadIdx.x;
    if (i < e) atomicAdd(&deg[dst[i]], 1.0f);
}

__global__ void k_dinv_inplace(float* __restrict__ d, int n) {
    int i = blockIdx.x * blockDim.x + threadIdx.x;
    if (i < n) d[i] = rsqrtf(d[i]);  // deg >= 1 always (self-loops)
}

__global__ void k_zero(float* __restrict__ p, int n) {
    int i = blockIdx.x * blockDim.x + threadIdx.x;
    if (i < n) p[i] = 0.0f;
}

// ---------------------------------------------------------------------------
// Layer 1 dense projection: hraw[N,4] = x[N,128] @ W1[128,4]
// WMMA f32 16x16x4, W1 zero-padded to 16 columns *inside LDS* so the B-operand
// read is a single unconditional ds_load_b64 (no exec-mask predication).
//
// LDS layout: sW[kk][n][j] = Wpad[2*kk + j][n], kk in [0,64), n in [0,16),
// j in {0,1}. For a K-step at base ka (even), lane (n = lane%16) reads the
// contiguous pair { Wpad[ka][n], Wpad[ka+1][n] } at sW + (kk*16 + n)*2.
//
// A layout (16x4 f32): lanes 0-15 hold {K=0,K=1}, lanes 16-31 hold {K=2,K=3}
// for row M = lane%16 -> one global_load_b64 per step (8B aligned, ka even).
// C/D: VGPR r holds row M=r (lanes<16) / M=8+r (lanes>=16), col N = lane%16.
// ---------------------------------------------------------------------------
__global__ __launch_bounds__(256) void k_gemm1_wmma(
    const float* __restrict__ x, const float* __restrict__ W1,
    float* __restrict__ hraw, int n) {
    __shared__ float sW[64 * 16 * 2];  // 8 KB padded/interleaved W1

    // Cooperative fill: element flat index -> (kk, nn, j)
    for (int idx = threadIdx.x; idx < 64 * 16 * 2; idx += blockDim.x) {
        const int j  = idx & 1;
        const int nn = (idx >> 1) & 15;
        const int kk = idx >> 5;
        sW[idx] = (nn < 4) ? W1[(2 * kk + j) * 4 + nn] : 0.0f;
    }
    __syncthreads();

    const int lane = threadIdx.x & 31;
    const int wave = threadIdx.x >> 5;
    const int half = lane >> 4;   // 0: K pair {k,k+1}; 1: K pair {k+2,k+3}
    const int l16  = lane & 15;   // A: row; B/C: col
    const int rowBase = blockIdx.x * 128 + wave * 16;

    int rowA = rowBase + l16;
    if (rowA >= n) rowA = n - 1;               // clamp: keep EXEC all-1s for WMMA
    const float* __restrict__ xr = x + (size_t)rowA * FEAT_IN;

    v8f c = {0.f, 0.f, 0.f, 0.f, 0.f, 0.f, 0.f, 0.f};

#pragma unroll
    for (int k = 0; k < FEAT_IN; k += 4) {
        const int ka = k + 2 * half;                       // even
        const v2f a = *(const v2f*)(xr + ka);              // global_load_b64
        const v2f b = *(const v2f*)(sW + ((ka >> 1) * 16 + l16) * 2);  // ds_load_b64
        c = __builtin_amdgcn_wmma_f32_16x16x4_f32(
            /*neg_a=*/false, a, /*neg_b=*/false, b,
            /*c_mod=*/(short)0, c, /*reuse_a=*/false, /*reuse_b=*/false);
    }

    const int m0 = rowBase + half * 8;
    if (l16 < 4) {
#pragma unroll
        for (int r = 0; r < 8; ++r) {
            const int m = m0 + r;
            if (m < n) hraw[(size_t)m * 4 + l16] = c[r];
        }
    }
}

// ---------------------------------------------------------------------------
// Small per-node dense layers (4x4, 4x2): register FMAs, bandwidth bound.
// ---------------------------------------------------------------------------
template <int FIN, int FOUT>
__global__ void k_node_linear(const float* __restrict__ h, const float* __restrict__ W,
                              float* __restrict__ out, int n) {
    int i = blockIdx.x * blockDim.x + threadIdx.x;
    if (i >= n) return;
    float in[FIN];
#pragma unroll
    for (int k = 0; k < FIN; ++k) in[k] = h[(size_t)i * FIN + k];
#pragma unroll
    for (int j = 0; j < FOUT; ++j) {
        float acc = 0.f;
#pragma unroll
        for (int k = 0; k < FIN; ++k) acc += in[k] * W[k * FOUT + j];
        out[(size_t)i * FOUT + j] = acc;
    }
}

// ---------------------------------------------------------------------------
// Edge scatter: agg[dst] += hraw[src] * dinv[src]*dinv[dst]   (self-loops
// folded into the post pass to save N*F atomics).
// ---------------------------------------------------------------------------
template <int F>
__global__ void k_scatter(const int* __restrict__ src, const int* __restrict__ dst,
                          const float* __restrict__ dinv, const float* __restrict__ hraw,
                          float* __restrict__ agg, int e) {
    int i = blockIdx.x * blockDim.x + threadIdx.x;
    if (i >= e) return;
    const int s = src[i];
    const int d = dst[i];
    const float nm = dinv[s] * dinv[d];
#pragma unroll
    for (int f = 0; f < F; ++f)
        atomicAdd(&agg[(size_t)d * F + f], hraw[(size_t)s * F + f] * nm);
}

// ---------------------------------------------------------------------------
// Post pass: out = tanh(agg + self_loop_term + bias)
// ---------------------------------------------------------------------------
template <int F>
__global__ void k_post_tanh(const float* __restrict__ agg, const float* __restrict__ hraw,
                            const float* __restrict__ dinv, const float* __restrict__ b,
                            float* __restrict__ out, int n) {
    int i = blockIdx.x * blockDim.x + threadIdx.x;
    if (i >= n) return;
    const float selfn = dinv[i] * dinv[i];
#pragma unroll
    for (int f = 0; f < F; ++f)
        out[(size_t)i * F + f] =
            tanhf(agg[(size_t)i * F + f] + hraw[(size_t)i * F + f] * selfn + b[f]);
}

// ---------------------------------------------------------------------------
// Classifier + output packing: d_out = [ out(N x 4) | h(N x 2) ]
// ---------------------------------------------------------------------------
__global__ void k_classify(const float* __restrict__ h, const float* __restrict__ Wc,
                           const float* __restrict__ bc, float* __restrict__ out, int n) {
    int i = blockIdx.x * blockDim.x + threadIdx.x;
    if (i >= n) return;
    const float h0 = h[(size_t)i * 2];
    const float h1 = h[(size_t)i * 2 + 1];
#pragma unroll
    for (int j = 0; j < 4; ++j)
        out[(size_t)i * 4 + j] = h0 * Wc[j] + h1 * Wc[4 + j] + bc[j];
    out[(size_t)n * 4 + (size_t)i * 2]     = h0;
    out[(size_t)n * 4 + (size_t)i * 2 + 1] = h1;
}

// ---------------------------------------------------------------------------
// Orchestration
// ---------------------------------------------------------------------------
extern "C" void kernel_launch(void* const* d_in, const int* in_sizes, int n_in,
                              void* d_out, int out_size, void* d_ws, size_t ws_size,
                              hipStream_t stream) {
    const float* x   = (const float*)d_in[0];
    const int*   ei  = (const int*)d_in[1];
    const float* W1  = (const float*)d_in[2];
    const float* b1  = (const float*)d_in[3];
    const float* W2  = (const float*)d_in[4];
    const float* b2  = (const float*)d_in[5];
    const float* W3  = (const float*)d_in[6];
    const float* b3  = (const float*)d_in[7];
    const float* Wc  = (const float*)d_in[8];
    const float* bc  = (const float*)d_in[9];
    float* out = (float*)d_out;

    const int n = N_NODES;
    const int e = in_sizes[1] / 2;        // 6,400,000
    const int* src = ei;                  // edge_index[0]
    const int* dst = ei + e;              // edge_index[1]

    // Workspace carve-up (floats): dinv[n] | hraw[4n] | agg[4n] | hcur[4n]
    float* ws   = (float*)d_ws;
    float* dinv = ws;
    float* hraw = ws + (size_t)n;
    float* agg  = ws + (size_t)5 * n;
    float* hcur = ws + (size_t)9 * n;

    const int T = 256;
    const int gN = (n + T - 1) / T;
    const int gE = (e + T - 1) / T;
    const int g4 = (4 * n + T - 1) / T;
    const int gW = (n + 127) / 128;       // gemm1: 128 rows per 256-thread block

    // --- normalization: dinv = rsqrt(1 + in_degree) ---
    k_init_deg<<<gN, T, 0, stream>>>(dinv, n);
    k_deg_acc<<<gE, T, 0, stream>>>(dst, dinv, e);
    k_dinv_inplace<<<gN, T, 0, stream>>>(dinv, n);

    // --- layer 1: WMMA projection, scatter, activation ---
    k_gemm1_wmma<<<gW, T, 0, stream>>>(x, W1, hraw, n);
    k_zero<<<g4, T, 0, stream>>>(agg, 4 * n);
    k_scatter<4><<<gE, T, 0, stream>>>(src, dst, dinv, hraw, agg, e);
    k_post_tanh<4><<<gN, T, 0, stream>>>(agg, hraw, dinv, b1, hcur, n);

    // --- layer 2: 4 -> 4 ---
    k_node_linear<4, 4><<<gN, T, 0, stream>>>(hcur, W2, hraw, n);
    k_zero<<<g4, T, 0, stream>>>(agg, 4 * n);
    k_scatter<4><<<gE, T, 0, stream>>>(src, dst, dinv, hraw, agg, e);
    k_post_tanh<4><<<gN, T, 0, stream>>>(agg, hraw, dinv, b2, hcur, n);

    // --- layer 3: 4 -> 2 ---
    k_node_linear<4, 2><<<gN, T, 0, stream>>>(hcur, W3, hraw, n);
    k_zero<<<g4, T, 0, stream>>>(agg, 2 * n);
    k_scatter<2><<<gE, T, 0, stream>>>(src, dst, dinv, hraw, agg, e);
    k_post_tanh<2><<<gN, T, 0, stream>>>(agg, hraw, dinv, b3, hcur, n);

    // --- classifier + output packing ---
    k_classify<<<gN, T, 0, stream>>>(hcur, Wc, bc, out, n);
}